// CombinedMarginLoss_63754494542030
// MI455X (gfx1250) — compile-verified
//
#include <hip/hip_runtime.h>
#include <math.h>
#include <stdint.h>

// ---------------- tunables ----------------
#define TILE   4000          // floats per TDM tile (16000 B); 100000 = 25 tiles
#define DEPTH  4             // TDM pipeline depth (4 LDS buffers, 3 in flight)
#define BLOCK  256           // threads per block (8 wave32)

typedef unsigned int v4u __attribute__((ext_vector_type(4)));
typedef int          v8i __attribute__((ext_vector_type(8)));
typedef int          v4i __attribute__((ext_vector_type(4)));

// ArcFace constants (m2 = 0.5)
#define COS_M2   0.8775825618903728f
#define SIN_M2   0.4794255386042030f
#define THRESH  (-0.8775825618903728f)   // cos(pi - 0.5)
#define SCALE_S  64.0f

// Issue one TDM DMA: copy `chunk` floats from gsrc into LDS at byte addr lds_byte.
// Descriptor per CDNA5 ISA 8.3/8.4: 1-D tile (tile_dim0 = chunk, tile_dim1 = 1).
__device__ __forceinline__ void tdm_load_tile(const float* gsrc,
                                              unsigned lds_byte, int chunk) {
  unsigned long long ga = (unsigned long long)(uintptr_t)gsrc;
  unsigned dim0 = (unsigned)chunk;

  v4u g0;
  g0[0] = 1u;                                        // count=1, user descriptor
  g0[1] = lds_byte;                                  // lds_addr
  g0[2] = (unsigned)(ga & 0xFFFFFFFFu);              // global_addr[31:0]
  g0[3] = (unsigned)((ga >> 32) & 0x1FFFFFFu)        // global_addr[56:32]
        | (2u << 30);                                // type = 2 ("image")

  v8i g1;
  g1[0] = (int)(2u << 16);                           // wg_mask=0, data_size=4B
  g1[1] = (int)((dim0 & 0xFFFFu) << 16);             // tensor_dim0[15:0]
  g1[2] = (int)((dim0 >> 16) | (1u << 16));          // tensor_dim0[31:16], tensor_dim1=1
  g1[3] = (int)((dim0 & 0xFFFFu) << 16);             // tensor_dim1 hi=0, tile_dim0=chunk
  g1[4] = 1;                                         // tile_dim1=1, tile_dim2=0
  g1[5] = (int)dim0;                                 // tensor_dim0_stride[31:0]
  g1[6] = 0;                                         // stride hi / dim1_stride lo
  g1[7] = 0;

  v4i g2 = {0, 0, 0, 0};                             // dims 2..4 unused
  v4i g3 = {0, 0, 0, 0};
  v8i g4 = {0, 0, 0, 0, 0, 0, 0, 0};                 // extra group (clang-23 form): zeros

  __builtin_amdgcn_tensor_load_to_lds(g0, g1, g2, g3, g4, /*cpol*/0);
}

// One block per row. Streams the row through LDS via TDM (DEPTH-deep pipeline),
// accumulates sum(exp(64*v - 64)) in fixed order, applies the ArcFace label
// correction, writes loss_b = 64 + log(sum) - 64*theta to row_loss[row].
__global__ __launch_bounds__(BLOCK)
void row_lse_kernel(const float* __restrict__ logits,
                    const int*   __restrict__ labels,
                    float*       __restrict__ row_loss,
                    int C) {
  __shared__ __align__(16) float bufs[DEPTH * TILE];
  __shared__ float warp_sums[BLOCK / 32];

  const int row = blockIdx.x;
  const int tid = threadIdx.x;
  const float* rowp = logits + (size_t)row * (size_t)C;
  const int NT = (C + TILE - 1) / TILE;

  const unsigned lds0 = (unsigned)(uintptr_t)&bufs[0];

  // Wave-0-only TDM issue. readfirstlane makes the condition provably wave-
  // uniform -> scalar branch, so waves 1..7 skip the tensor ops entirely
  // (TDM instructions ignore EXEC, so a VALU mask would not be enough).
  const int waveid = __builtin_amdgcn_readfirstlane((int)(threadIdx.x >> 5));

  if (waveid == 0) {
    const int pre = (NT < (DEPTH - 1)) ? NT : (DEPTH - 1);
    for (int t = 0; t < pre; ++t) {
      int chunk = min(TILE, C - t * TILE);
      tdm_load_tile(rowp + t * TILE, lds0 + (unsigned)(t % DEPTH) * (TILE * 4u), chunk);
    }
  }

  float s = 0.0f;
  for (int t = 0; t < NT; ++t) {
    if (waveid == 0) {
      const int n = t + (DEPTH - 1);
      if (n < NT) {
        int chunk = min(TILE, C - n * TILE);
        tdm_load_tile(rowp + n * TILE, lds0 + (unsigned)(n % DEPTH) * (TILE * 4u), chunk);
        __builtin_amdgcn_s_wait_tensorcnt(DEPTH - 1);  // oldest (tile t) done
      } else {
        const int pend = NT - 1 - t;                   // tiles still in flight past t
        if (pend <= 0)      __builtin_amdgcn_s_wait_tensorcnt(0);
        else if (pend == 1) __builtin_amdgcn_s_wait_tensorcnt(1);
        else                __builtin_amdgcn_s_wait_tensorcnt(2);
      }
    }
    __syncthreads();                                   // tile t visible to all waves

    const float* buf = &bufs[(t % DEPTH) * TILE];
    const int chunk = min(TILE, C - t * TILE);
    const int c4 = chunk >> 2;
    const float4* bv = (const float4*)buf;
    for (int i = tid; i < c4; i += BLOCK) {            // ds_load_b128 per iter
      float4 v = bv[i];
      s += __expf(SCALE_S * v.x - SCALE_S);
      s += __expf(SCALE_S * v.y - SCALE_S);
      s += __expf(SCALE_S * v.z - SCALE_S);
      s += __expf(SCALE_S * v.w - SCALE_S);
    }
    for (int i = (c4 << 2) + tid; i < chunk; i += BLOCK) {
      s += __expf(SCALE_S * buf[i] - SCALE_S);
    }
    __syncthreads();                                   // reads done before slot reuse
  }

  // wave32 shuffle reduce, then cross-wave via LDS (fixed order -> deterministic)
  #pragma unroll
  for (int off = 16; off > 0; off >>= 1) s += __shfl_down(s, off, 32);
  if ((tid & 31) == 0) warp_sums[tid >> 5] = s;
  __syncthreads();

  if (tid == 0) {
    float tot = 0.0f;
    #pragma unroll
    for (int w = 0; w < BLOCK / 32; ++w) tot += warp_sums[w];

    const int   label  = labels[row];
    const float target = rowp[label];
    const float tc     = fminf(fmaxf(target, -1.0f + 1e-7f), 1.0f - 1e-7f);
    // cos(acos(tc) + m2) = tc*cos(m2) - sqrt(1-tc^2)*sin(m2)
    const float cos_m  = tc * COS_M2 - sqrtf(fmaxf(1.0f - tc * tc, 0.0f)) * SIN_M2;
    const float theta  = (target > THRESH) ? cos_m : (-2.0f - cos_m);
    const float xl     = SCALE_S * theta;

    // swap label term: remove exp(64*t - 64), add exp(64*theta - 64)
    tot += __expf(xl - SCALE_S) - __expf(SCALE_S * target - SCALE_S);

    // loss_b = logsumexp - x_label, with fixed shift 64 (all margin logits <= 64)
    row_loss[row] = SCALE_S + logf(tot) - xl;
  }
}

// Deterministic mean over B per-row losses -> d_out[0].
__global__ void mean_kernel(const float* __restrict__ row_loss,
                            float* __restrict__ out, int B) {
  __shared__ float wsum[32];
  const int tid = threadIdx.x;
  float s = 0.0f;
  for (int i = tid; i < B; i += (int)blockDim.x) s += row_loss[i];
  #pragma unroll
  for (int off = 16; off > 0; off >>= 1) s += __shfl_down(s, off, 32);
  if ((tid & 31) == 0) wsum[tid >> 5] = s;
  __syncthreads();
  if (tid == 0) {
    float tot = 0.0f;
    const int nw = ((int)blockDim.x + 31) / 32;
    for (int w = 0; w < nw; ++w) tot += wsum[w];
    out[0] = tot / (float)B;
  }
}

extern "C" void kernel_launch(void* const* d_in, const int* in_sizes, int n_in,
                              void* d_out, int out_size, void* d_ws, size_t ws_size,
                              hipStream_t stream) {
  const float* logits = (const float*)d_in[0];
  const int*   labels = (const int*)d_in[1];
  const int B = in_sizes[1];
  const int C = in_sizes[0] / B;

  float* row_loss = (float*)d_ws;   // B floats of scratch

  row_lse_kernel<<<B, BLOCK, 0, stream>>>(logits, labels, row_loss, C);
  mean_kernel<<<1, 512, 0, stream>>>(row_loss, (float*)d_out, B);
}